// GCN_747324309860
// MI455X (gfx1250) — compile-verified
//
#include <hip/hip_runtime.h>

typedef float v2f __attribute__((ext_vector_type(2)));
typedef float v8f __attribute__((ext_vector_type(8)));

#define HID 128
#define NUM_GRAPHS 512
#define GN_EPS 1e-5f

// ---------------- degree / normalization ----------------
__global__ void k_dis_init(float* dis, int N) {
    int i = blockIdx.x * blockDim.x + threadIdx.x;
    if (i < N) dis[i] = 1.0f;  // self-loop contributes 1 to deg
}

__global__ void k_deg_edges(float* dis, const int* __restrict__ dst, int E) {
    int e = blockIdx.x * blockDim.x + threadIdx.x;
    if (e < E) atomicAdd(&dis[dst[e]], 1.0f);
}

__global__ void k_dis_finish(float* dis, int N) {
    int i = blockIdx.x * blockDim.x + threadIdx.x;
    if (i < N) dis[i] = rsqrtf(dis[i]);
}

// ---------------- graph segment starts (batch is sorted) ----------------
__global__ void k_starts(const int* __restrict__ batch, int* __restrict__ start, int N) {
    int g = blockIdx.x * blockDim.x + threadIdx.x;
    if (g > NUM_GRAPHS) return;
    if (g == NUM_GRAPHS) { start[g] = N; return; }
    int lo = 0, hi = N;                 // first i with batch[i] >= g
    while (lo < hi) {
        int mid = (lo + hi) >> 1;
        if (batch[mid] < g) lo = mid + 1; else hi = mid;
    }
    start[g] = lo;
}

// ---------------- WMMA f32 GEMM: C[N x 128] = X[N x K] @ W[K x 128] (+bias) ----------------
// Block = 256 threads (8 wave32). Block owns one 16-row M tile; wave w owns N-tile w.
// A staged through LDS; B fragments read from global (W is hot in cache).
// CONTRACT: N must be a multiple of 16 (true for 100000 and 512).
template <int K>
__global__ void k_gemm_wmma(const float* __restrict__ X, const float* __restrict__ W,
                            const float* __restrict__ bias, float* __restrict__ C, int N) {
    __shared__ float lds_a[16 * (K + 2)];
    const int tid = threadIdx.x;
    const int mt  = blockIdx.x;

    // stage A tile [16 x K] (coalesced); no tail: N % 16 == 0
    for (int e = tid; e < 16 * K; e += 256) {
        int r = e / K, c = e - r * K;
        lds_a[r * (K + 2) + c] = X[(size_t)(mt * 16 + r) * K + c];
    }
    __syncthreads();

    const int nt    = tid >> 5;        // wave id -> N tile (0..7)
    const int lane  = tid & 31;
    const int m_loc = lane & 15;       // also n_loc for B/C
    const int kh    = lane >> 4;       // lane half selects K pair / M+8 rows

    v8f acc = {0.f, 0.f, 0.f, 0.f, 0.f, 0.f, 0.f, 0.f};
    const float* arow = &lds_a[m_loc * (K + 2)];
    const int ncol = nt * 16 + m_loc;

#pragma unroll
    for (int k0 = 0; k0 < K; k0 += 4) {
        // A 16x4 f32: VGPR0 = K(0|2), VGPR1 = K(1|3) by lane half
        v2f a;
        a.x = arow[k0 + 2 * kh];
        a.y = arow[k0 + 2 * kh + 1];
        // B 4x16 f32: VGPR0 = rows (0|2), VGPR1 = rows (1|3) by lane half
        v2f b;
        b.x = W[(size_t)(k0 + 2 * kh) * HID + ncol];
        b.y = W[(size_t)(k0 + 2 * kh + 1) * HID + ncol];
        acc = __builtin_amdgcn_wmma_f32_16x16x4_f32(false, a, false, b, (short)0, acc,
                                                    false, false);
    }

    // C/D layout: VGPR v -> row v + 8*kh, col = lane&15
    const float bv = bias ? bias[ncol] : 0.0f;   // loop-invariant, hoisted
#pragma unroll
    for (int v = 0; v < 8; ++v) {
        int row = mt * 16 + v + 8 * kh;
        C[(size_t)row * HID + ncol] = acc[v] + bv;
    }
}

// ---------------- GCN aggregation ----------------
// init: h = bias + dis^2 * xw   (self-loop term, atomic-free full overwrite)
__global__ void k_selfloop_bias(const float* __restrict__ xw, const float* __restrict__ dis,
                                const float* __restrict__ bias, float* __restrict__ h, int N) {
    size_t i = (size_t)blockIdx.x * blockDim.x + threadIdx.x;
    if (i >= (size_t)N * HID) return;
    int node = (int)(i >> 7);
    int c    = (int)(i & (HID - 1));
    float dn = dis[node];
    h[i] = bias[c] + dn * dn * xw[i];
}

// one wave per edge: h[dst] += dis[src]*dis[dst] * xw[src]
__global__ void k_scatter_edges(const float* __restrict__ xw, const int* __restrict__ src,
                                const int* __restrict__ dst, const float* __restrict__ dis,
                                float* __restrict__ h, int E) {
    int wid  = (int)(((size_t)blockIdx.x * blockDim.x + threadIdx.x) >> 5);
    int lane = threadIdx.x & 31;
    if (wid >= E) return;
    int s = src[wid], d = dst[wid];
    float norm = dis[s] * dis[d];
    const float4 v = ((const float4*)(xw + (size_t)s * HID))[lane];
    float* hd = h + (size_t)d * HID + lane * 4;
    atomicAdd(hd + 0, norm * v.x);
    atomicAdd(hd + 1, norm * v.y);
    atomicAdd(hd + 2, norm * v.z);
    atomicAdd(hd + 3, norm * v.w);
}

// ---------------- GraphNorm (in-place), one block (128 thr) per graph ----------------
__global__ void k_graphnorm(float* __restrict__ h, const float* __restrict__ w,
                            const float* __restrict__ b, const float* __restrict__ s,
                            const int* __restrict__ start, int relu) {
    int g = blockIdx.x, c = threadIdx.x;
    int s0 = start[g], s1 = start[g + 1];
    int n = s1 - s0;
    if (n <= 0) return;
    float inv = 1.0f / (float)n;

    float sum = 0.0f;
    for (int i = s0; i < s1; ++i) sum += h[(size_t)i * HID + c];
    float mean_s = sum * inv * s[c];            // fold mean_scale into mean

    float var = 0.0f;
    for (int i = s0; i < s1; ++i) {
        float d = h[(size_t)i * HID + c] - mean_s;
        var += d * d;
    }
    float rs = rsqrtf(var * inv + GN_EPS);
    float wc = w[c], bc = b[c];
    for (int i = s0; i < s1; ++i) {
        float d = h[(size_t)i * HID + c] - mean_s;
        float o = wc * d * rs + bc;
        if (relu) o = fmaxf(o, 0.0f);
        h[(size_t)i * HID + c] = o;
    }
}

// ---------------- global mean pool ----------------
__global__ void k_meanpool(const float* __restrict__ h, const int* __restrict__ start,
                           float* __restrict__ pool) {
    int g = blockIdx.x, c = threadIdx.x;
    int s0 = start[g], s1 = start[g + 1];
    int n = s1 - s0;
    float sum = 0.0f;
    for (int i = s0; i < s1; ++i) sum += h[(size_t)i * HID + c];
    pool[(size_t)g * HID + c] = (n > 0) ? sum / (float)n : 0.0f;
}

// ---------------- final 128 -> 2 linear ----------------
__global__ void k_lin3(const float* __restrict__ gin, const float* __restrict__ W,
                       const float* __restrict__ b, float* __restrict__ out) {
    int idx = blockIdx.x * blockDim.x + threadIdx.x;   // 512*2
    if (idx >= NUM_GRAPHS * 2) return;
    int g = idx >> 1, o = idx & 1;
    float sum = b[o];
#pragma unroll 8
    for (int k = 0; k < HID; ++k) sum += gin[(size_t)g * HID + k] * W[k * 2 + o];
    out[idx] = sum;
}

static inline size_t align256(size_t x) { return (x + 255) & ~(size_t)255; }

extern "C" void kernel_launch(void* const* d_in, const int* in_sizes, int n_in,
                              void* d_out, int out_size, void* d_ws, size_t ws_size,
                              hipStream_t stream) {
    const float* x     = (const float*)d_in[0];
    const int*   ei    = (const int*)d_in[1];     // [2, E]: row0=src, row1=dst
    const int*   batch = (const int*)d_in[2];
    const float* W1 = (const float*)d_in[3];  const float* b1 = (const float*)d_in[4];
    const float* W2 = (const float*)d_in[5];  const float* b2 = (const float*)d_in[6];
    const float* W3 = (const float*)d_in[7];  const float* b3 = (const float*)d_in[8];
    const float* gn1w = (const float*)d_in[9];  const float* gn1b = (const float*)d_in[10]; const float* gn1s = (const float*)d_in[11];
    const float* gn2w = (const float*)d_in[12]; const float* gn2b = (const float*)d_in[13]; const float* gn2s = (const float*)d_in[14];
    const float* gn3w = (const float*)d_in[15]; const float* gn3b = (const float*)d_in[16]; const float* gn3s = (const float*)d_in[17];
    const float* l1W = (const float*)d_in[18]; const float* l1b = (const float*)d_in[19];
    const float* l2W = (const float*)d_in[20]; const float* l2b = (const float*)d_in[21];
    const float* l3W = (const float*)d_in[22]; const float* l3b = (const float*)d_in[23];

    const int N = in_sizes[0] / 64;   // 100000
    const int E = in_sizes[1] / 2;    // 1600000
    const int* src = ei;
    const int* dst = ei + E;

    // workspace carve-up
    char* p = (char*)d_ws;
    float* dis   = (float*)p; p += align256((size_t)N * 4);
    int*   start = (int*)p;   p += align256((NUM_GRAPHS + 1) * 4);
    float* bufA  = (float*)p; p += align256((size_t)N * HID * 4);
    float* bufB  = (float*)p; p += align256((size_t)N * HID * 4);
    float* pool  = (float*)p; p += align256((size_t)NUM_GRAPHS * HID * 4);
    float* gtmp  = (float*)p; p += align256((size_t)NUM_GRAPHS * HID * 4);

    const int T = 256;
    const int nBlkN   = (N + T - 1) / T;
    const int nBlkE   = (E + T - 1) / T;
    const int nBlkNC  = (int)(((size_t)N * HID + T - 1) / T);
    const int nBlkEW  = (E * 32 + T - 1) / T;   // one wave per edge
    const int nTilesN = (N + 15) / 16;          // 6250 (exact)
    const int nTilesG = (NUM_GRAPHS + 15) / 16; // 32 (exact)

    // degree -> dis = rsqrt(deg)
    k_dis_init  <<<nBlkN, T, 0, stream>>>(dis, N);
    k_deg_edges <<<nBlkE, T, 0, stream>>>(dis, dst, E);
    k_dis_finish<<<nBlkN, T, 0, stream>>>(dis, N);
    k_starts    <<<3, T, 0, stream>>>(batch, start, N);

    // ---- layer 1: x[N,64] @ W1 -> aggregate -> GraphNorm+ReLU
    k_gemm_wmma<64><<<nTilesN, T, 0, stream>>>(x, W1, nullptr, bufA, N);
    k_selfloop_bias<<<nBlkNC, T, 0, stream>>>(bufA, dis, b1, bufB, N);
    k_scatter_edges<<<nBlkEW, T, 0, stream>>>(bufA, src, dst, dis, bufB, E);
    k_graphnorm<<<NUM_GRAPHS, HID, 0, stream>>>(bufB, gn1w, gn1b, gn1s, start, 1);

    // ---- layer 2
    k_gemm_wmma<128><<<nTilesN, T, 0, stream>>>(bufB, W2, nullptr, bufA, N);
    k_selfloop_bias<<<nBlkNC, T, 0, stream>>>(bufA, dis, b2, bufB, N);
    k_scatter_edges<<<nBlkEW, T, 0, stream>>>(bufA, src, dst, dis, bufB, E);
    k_graphnorm<<<NUM_GRAPHS, HID, 0, stream>>>(bufB, gn2w, gn2b, gn2s, start, 1);

    // ---- layer 3 (no ReLU)
    k_gemm_wmma<128><<<nTilesN, T, 0, stream>>>(bufB, W3, nullptr, bufA, N);
    k_selfloop_bias<<<nBlkNC, T, 0, stream>>>(bufA, dis, b3, bufB, N);
    k_scatter_edges<<<nBlkEW, T, 0, stream>>>(bufA, src, dst, dis, bufB, E);
    k_graphnorm<<<NUM_GRAPHS, HID, 0, stream>>>(bufB, gn3w, gn3b, gn3s, start, 0);

    // ---- pool + head
    k_meanpool<<<NUM_GRAPHS, HID, 0, stream>>>(bufB, start, pool);
    k_gemm_wmma<128><<<nTilesG, T, 0, stream>>>(pool, l1W, l1b, gtmp, NUM_GRAPHS);
    k_gemm_wmma<128><<<nTilesG, T, 0, stream>>>(gtmp, l2W, l2b, pool, NUM_GRAPHS);
    k_lin3<<<(NUM_GRAPHS * 2 + T - 1) / T, T, 0, stream>>>(pool, l3W, l3b, (float*)d_out);
}